// Matching_loss_61117384622877
// MI455X (gfx1250) — compile-verified
//
#include <hip/hip_runtime.h>
#include <hip/hip_bf16.h>
#include <math.h>

// Problem constants from the reference
#define BB 4
#define CC 32
#define MM 1024
#define INV_T 10.0f          // (1/dist)/T with T = 0.1  ->  10/dist

typedef __attribute__((ext_vector_type(2))) float v2f;
typedef __attribute__((ext_vector_type(8))) float v8f;

// ---------------- block reduction helpers (256 threads) ----------------
__device__ __forceinline__ float block_reduce_sum(float v, float* sm) {
    int tid = threadIdx.x;
    sm[tid] = v;
    __syncthreads();
    for (int s = 128; s > 0; s >>= 1) {
        if (tid < s) sm[tid] += sm[tid + s];
        __syncthreads();
    }
    float r = sm[0];
    __syncthreads();
    return r;
}

__device__ __forceinline__ float block_reduce_max(float v, float* sm) {
    int tid = threadIdx.x;
    sm[tid] = v;
    __syncthreads();
    for (int s = 128; s > 0; s >>= 1) {
        if (tid < s) sm[tid] = fmaxf(sm[tid], sm[tid + s]);
        __syncthreads();
    }
    float r = sm[0];
    __syncthreads();
    return r;
}

// ---------------- kernel: zero the scalar output ----------------
__global__ void zero_out_k(float* out) {
    if (threadIdx.x == 0 && blockIdx.x == 0) out[0] = 0.0f;
}

// ---------------- kernel: squared column norms of both descriptor sets ----
// grid: 32 blocks x 256 threads = 8192 = 2 * B * M
__global__ void norms_k(const float* __restrict__ srcd, const float* __restrict__ dstd,
                        float* __restrict__ ns, float* __restrict__ nd) {
    int idx   = blockIdx.x * blockDim.x + threadIdx.x;   // 0..8191
    int which = idx >> 12;                               // 0 = src, 1 = dst
    int r     = idx & 4095;                              // b*M + i
    int b     = r >> 10;
    int i     = r & (MM - 1);
    const float* p = (which ? dstd : srcd) + (size_t)b * CC * MM + i;
    float s = 0.0f;
#pragma unroll
    for (int c = 0; c < CC; ++c) {
        float v = p[(size_t)c * MM];
        s += v * v;
    }
    (which ? nd : ns)[r] = s;
}

// ---------------- kernel: normalized sigma weights ----------------
// grid: B blocks x 256 threads; w = clip(3 - sigma, min 0.01) / mean(w)
__global__ void weights_k(const float* __restrict__ sigma, float* __restrict__ w) {
    __shared__ float sm[256];
    int b   = blockIdx.x;
    int tid = threadIdx.x;
    const float* sg = sigma + (size_t)b * MM;
    float vals[4];
    float p = 0.0f;
#pragma unroll
    for (int t = 0; t < 4; ++t) {
        vals[t] = fmaxf(3.0f - sg[t * 256 + tid], 0.01f);
        p += vals[t];
    }
    float total    = block_reduce_sum(p, sm);
    float inv_mean = (float)MM / total;
#pragma unroll
    for (int t = 0; t < 4; ++t)
        w[(size_t)b * MM + t * 256 + tid] = vals[t] * inv_mean;
}

// ---------------- kernel: WMMA Gram matrix -> logits ----------------
// Each wave computes one 16x16 tile of S = src_desc^T * dst_desc via
// v_wmma_f32_16x16x4_f32, then logit = (1/dist)/T with
// dist^2 = ||s_i||^2 + ||d_j||^2 - 2 * dot(i,j).
// grid.x = B * 64 (i-tiles) * 8 (j-groups of 8 tiles); block = 256 (8 waves).
__global__ void gram_logits_k(const float* __restrict__ srcd, const float* __restrict__ dstd,
                              const float* __restrict__ ns, const float* __restrict__ nd,
                              float* __restrict__ L) {
    int blk = blockIdx.x;
    int jg  = blk & 7;  blk >>= 3;
    int ti  = blk & 63; blk >>= 6;
    int b   = blk;                         // 0..3
    int wave = threadIdx.x >> 5;
    int lane = threadIdx.x & 31;
    int tj   = jg * 8 + wave;              // 0..63
    int i0   = ti * 16;
    int j0   = tj * 16;
    int half = lane >> 4;                  // 0 or 1
    int l    = lane & 15;

    const float* S = srcd + (size_t)b * CC * MM;
    const float* D = dstd + (size_t)b * CC * MM;

    v8f acc = {};
    // A 16x4 layout: v0 = K=(0|2), v1 = K=(1|3)  (halves of the wave)
    // B 4x16 layout mirrored. K-loop: 32 = 8 x (K=4) WMMA ops.
#pragma unroll
    for (int k0 = 0; k0 < CC; k0 += 4) {
        int ka = k0 + 2 * half;
        v2f av, bv;
        av.x = S[(size_t)(ka + 0) * MM + i0 + l];
        av.y = S[(size_t)(ka + 1) * MM + i0 + l];
        bv.x = D[(size_t)(ka + 0) * MM + j0 + l];
        bv.y = D[(size_t)(ka + 1) * MM + j0 + l];
        acc = __builtin_amdgcn_wmma_f32_16x16x4_f32(
            /*neg_a=*/false, av, /*neg_b=*/false, bv,
            /*c_mod=*/(short)0, acc, /*reuse_a=*/false, /*reuse_b=*/false);
    }

    // C/D layout: VGPR r -> row (half ? r+8 : r), lane l -> col l
    const float* nsb = ns + (size_t)b * MM;
    const float* ndb = nd + (size_t)b * MM;
    float* Lb = L + (size_t)b * MM * MM;
#pragma unroll
    for (int r = 0; r < 8; ++r) {
        int gi = i0 + (half ? r + 8 : r);
        int gj = j0 + l;
        float d2 = nsb[gi] + ndb[gj] - 2.0f * acc[r];
        d2 = fmaxf(d2, 1e-12f);
        float dist = sqrtf(d2);
        Lb[(size_t)gi * MM + gj] = INV_T / dist;
    }
}

// ---------------- kernel: softmax + keypoint correspondence + loss ------
// One block per (b, r). transposed=0: r = src row i, softmax over j (dst),
// kp_other = dst_kp, kp_self = src_kp, w = src_w.
// transposed=1: r = dst col j, softmax over i (strided column read).
// grid = B*M = 4096 blocks x 256 threads.
__global__ void softmax_corr_k(const float* __restrict__ L,
                               const float* __restrict__ kp_self,
                               const float* __restrict__ kp_other,
                               const float* __restrict__ w,
                               float* __restrict__ out,
                               int transposed) {
    __shared__ float sm[256];
    int tid = threadIdx.x;
    int b   = blockIdx.x >> 10;
    int r   = blockIdx.x & (MM - 1);
    const float* Lb = L + (size_t)b * MM * MM;
    const float* ko = kp_other + (size_t)b * 3 * MM;

    float v[4], ox[4], oy[4], oz[4];
#pragma unroll
    for (int t = 0; t < 4; ++t) {
        int idx = t * 256 + tid;
        v[t]  = transposed ? Lb[(size_t)idx * MM + r] : Lb[(size_t)r * MM + idx];
        ox[t] = ko[idx];
        oy[t] = ko[MM + idx];
        oz[t] = ko[2 * MM + idx];
    }
    float m = fmaxf(fmaxf(v[0], v[1]), fmaxf(v[2], v[3]));
    m = block_reduce_max(m, sm);

    float ps = 0.0f, px = 0.0f, py = 0.0f, pz = 0.0f;
#pragma unroll
    for (int t = 0; t < 4; ++t) {
        float e = expf(v[t] - m);
        ps += e;
        px += e * ox[t];
        py += e * oy[t];
        pz += e * oz[t];
    }
    ps = block_reduce_sum(ps, sm);
    px = block_reduce_sum(px, sm);
    py = block_reduce_sum(py, sm);
    pz = block_reduce_sum(pz, sm);

    if (tid == 0) {
        float inv = 1.0f / ps;
        const float* ks = kp_self + (size_t)b * 3 * MM;
        float dx = ks[r]          - px * inv;
        float dy = ks[MM + r]     - py * inv;
        float dz = ks[2 * MM + r] - pz * inv;
        float diff = sqrtf(dx * dx + dy * dy + dz * dz);
        atomicAdd(out, w[(size_t)b * MM + r] * diff * (1.0f / (float)(BB * MM)));
    }
}

// ---------------- host launch ----------------
extern "C" void kernel_launch(void* const* d_in, const int* in_sizes, int n_in,
                              void* d_out, int out_size, void* d_ws, size_t ws_size,
                              hipStream_t stream) {
    const float* src_kp    = (const float*)d_in[0];  // [B,3,M]
    const float* src_sigma = (const float*)d_in[1];  // [B,M]
    const float* src_desc  = (const float*)d_in[2];  // [B,C,M]
    const float* dst_kp    = (const float*)d_in[3];  // [B,3,M]
    const float* dst_sigma = (const float*)d_in[4];  // [B,M]
    const float* dst_desc  = (const float*)d_in[5];  // [B,C,M]
    float* out = (float*)d_out;

    float* ws = (float*)d_ws;
    float* L  = ws;                                  // B*M*M floats (16 MB)
    float* ns = ws + (size_t)BB * MM * MM;           // B*M
    float* nd = ns + (size_t)BB * MM;                // B*M
    float* sw = nd + (size_t)BB * MM;                // B*M
    float* dw = sw + (size_t)BB * MM;                // B*M

    zero_out_k<<<1, 32, 0, stream>>>(out);
    norms_k<<<32, 256, 0, stream>>>(src_desc, dst_desc, ns, nd);
    weights_k<<<BB, 256, 0, stream>>>(src_sigma, sw);
    weights_k<<<BB, 256, 0, stream>>>(dst_sigma, dw);
    gram_logits_k<<<BB * 64 * 8, 256, 0, stream>>>(src_desc, dst_desc, ns, nd, L);
    // forward: rows (softmax over dst j), self = src_kp, other = dst_kp
    softmax_corr_k<<<BB * MM, 256, 0, stream>>>(L, src_kp, dst_kp, sw, out, 0);
    // backward: columns (softmax over src i), self = dst_kp, other = src_kp
    softmax_corr_k<<<BB * MM, 256, 0, stream>>>(L, dst_kp, src_kp, dw, out, 1);
}